// NetGCN_53609781789204
// MI455X (gfx1250) — compile-verified
//
#include <hip/hip_runtime.h>
#include <hip/hip_bf16.h>

// ---------------------------------------------------------------------------
// GCN pipeline for MI455X (gfx1250, wave32).
//  - Dense node GEMMs use V_WMMA_F32_16X16X4_F32 (fp32 in / fp32 acc; problem
//    is memory/scatter bound at ~2.7 GFLOP vs 64 MB HBM traffic, so fp32 WMMA
//    keeps reference numerics at zero cost).
//  - Weights staged in LDS in *fragment order* so each lane's B fragment is a
//    single ds_load_b64; A fragment is one global_load_b64 shared across all
//    N-tiles (accumulators held in registers); A loads software-pipelined.
//  - BN statistics fused into the GEMM via LDS partials + global f32 atomics.
//  - Edge aggregation: float4 reads + global_atomic_add_f32 scatter (L2-
//    resident working set, 192MB L2).
//  - Segment max pooling: monotonic u32 encoding + atomicMax.
// ---------------------------------------------------------------------------

typedef __attribute__((ext_vector_type(2))) float v2f;
typedef __attribute__((ext_vector_type(8))) float v8f;

#define EPS 1e-5f
#define ENC_NINF 0x007FFFFFu   // encf(-inf)

__device__ __forceinline__ unsigned encf(float x) {
    unsigned u = __float_as_uint(x);
    return (u & 0x80000000u) ? ~u : (u | 0x80000000u);
}
__device__ __forceinline__ float decf(unsigned u) {
    float v = (u & 0x80000000u) ? __uint_as_float(u & 0x7FFFFFFFu)
                                : __uint_as_float(~u);
    return isfinite(v) ? v : 0.0f;
}

// ---------------- fill kernels ----------------
__global__ void fill_f32_kernel(float* p, long n, float v) {
    long i = (long)blockIdx.x * blockDim.x + threadIdx.x;
    if (i < n) p[i] = v;
}
__global__ void fill_u32_kernel(unsigned* p, long n, unsigned v) {
    long i = (long)blockIdx.x * blockDim.x + threadIdx.x;
    if (i < n) p[i] = v;
}

// ---------------- WMMA GEMM: Y[N,NCOL] = X[N,K] @ W[K,NCOL] (+bias) ------
// One wave per 16-node tile; fully-unrolled K-loop of V_WMMA_F32_16X16X4_F32.
// A frag (16x4):  lanes 0-15 = rows, VGPRs = K(k,k+1); lanes 16-31 = K(k+2,k+3)
// B frag (4x16):  lanes 0-15 = cols, VGPRs = K(k,k+1); lanes 16-31 = K(k+2,k+3)
// C/D (16x16):    VGPR r = row r (lanes 0-15) / row r+8 (lanes 16-31), lane%16 = col
// W is staged in LDS as K/2 x NCOL array of (W[2t][c], W[2t+1][c]) pairs so a
// lane's B fragment is one aligned ds_load_b64.
// Optionally accumulates per-column sum/sumsq (BatchNorm) into stats[2*NCOL].
template <int K, int NCOL>
__global__ void gemm_wmma_kernel(const float* __restrict__ X,
                                 const float* __restrict__ W,
                                 const float* __restrict__ bias,
                                 float* __restrict__ Y,
                                 float* __restrict__ stats, int N) {
    constexpr int NT = NCOL / 16;
    extern __shared__ float lds[];
    float* ssum = lds;                   // [64] per-block BN partial sums
    float* ssq  = lds + 64;              // [64]
    v2f*   ldsW = (v2f*)(lds + 128);     // [(K/2)*NCOL] B-fragment pairs

    const int tid = threadIdx.x;
    if (stats) {
        for (int i = tid; i < 128; i += blockDim.x) lds[i] = 0.0f;
    }
    for (int i = tid; i < (K / 2) * NCOL; i += blockDim.x) {
        const int t = i / NCOL, col = i % NCOL;
        v2f w;
        w[0] = W[(2 * t) * NCOL + col];
        w[1] = W[(2 * t + 1) * NCOL + col];
        ldsW[i] = w;
    }
    __syncthreads();

    const int lane    = tid & 31;
    const int wave    = tid >> 5;
    const int tile    = blockIdx.x * (blockDim.x >> 5) + wave;
    const int row16   = tile << 4;
    const int halfsel = lane >> 4;       // 0: lanes 0-15, 1: lanes 16-31
    const int l16     = lane & 15;

    if (row16 < N) {
        const int arow = min(row16 + l16, N - 1);
        // lane-resolved pointers: A starts at K-offset 2*halfsel
        const float* __restrict__ xr = X + (long)arow * K + 2 * halfsel;
        const v2f* __restrict__ wfrag = ldsW + halfsel * NCOL + l16;

        v8f c[NT];
#pragma unroll
        for (int j = 0; j < NT; ++j) {
            const float bval = bias ? bias[j * 16 + l16] : 0.0f;
            c[j] = v8f{bval, bval, bval, bval, bval, bval, bval, bval};
        }

        v2f a = *(const v2f*)xr;         // software-pipelined A fragment
#pragma unroll
        for (int k = 0; k < K; k += 4) {
            v2f an = a;
            if (k + 4 < K) an = *(const v2f*)(xr + k + 4);
#pragma unroll
            for (int j = 0; j < NT; ++j) {
                const v2f b = wfrag[(k >> 1) * NCOL + j * 16];
                c[j] = __builtin_amdgcn_wmma_f32_16x16x4_f32(
                           /*neg_a=*/false, a, /*neg_b=*/false, b,
                           /*c_mod=*/(short)0, c[j],
                           /*reuse_a=*/false, /*reuse_b=*/false);
            }
            a = an;
        }

#pragma unroll
        for (int j = 0; j < NT; ++j) {
            float ps = 0.0f, pq = 0.0f;
#pragma unroll
            for (int r = 0; r < 8; ++r) {
                const int m = row16 + r + (halfsel << 3);
                if (m < N) Y[(long)m * NCOL + j * 16 + l16] = c[j][r];
                ps += c[j][r];
                pq += c[j][r] * c[j][r];
            }
            if (stats) {
                atomicAdd(&ssum[j * 16 + l16], ps);
                atomicAdd(&ssq[j * 16 + l16], pq);
            }
        }
    }
    if (stats) {
        __syncthreads();
        for (int i = tid; i < NCOL; i += blockDim.x) {
            atomicAdd(&stats[i], ssum[i]);
            atomicAdd(&stats[NCOL + i], ssq[i]);
        }
    }
}

// ---------------- BatchNorm finalize: mu, rsqrt(var+eps) ------------------
__global__ void finalize_stats_kernel(const float* __restrict__ s,
                                      float* __restrict__ o, float invN) {
    int c = threadIdx.x;                 // 32 threads
    float mu  = s[c] * invN;
    float var = s[32 + c] * invN - mu * mu;
    o[c]      = mu;
    o[32 + c] = rsqrtf(var + EPS);
}

// ---------------- BN + ReLU elementwise (C == 32) -------------------------
__global__ void bn_relu_kernel(float* __restrict__ h,
                               const float* __restrict__ murs,
                               const float* __restrict__ gamma,
                               const float* __restrict__ beta, long total) {
    long i = (long)blockIdx.x * blockDim.x + threadIdx.x;
    if (i >= total) return;
    int c = (int)(i & 31);
    float v = gamma[c] * (h[i] - murs[c]) * murs[32 + c] + beta[c];
    h[i] = v > 0.0f ? v : 0.0f;
}

// ---------------- initial_linear + segment-max pool -----------------------
__global__ void pool0_kernel(const float* __restrict__ h,
                             const float* __restrict__ Wl0,
                             const float* __restrict__ bl0,
                             const int* __restrict__ batch,
                             unsigned* __restrict__ pool0, int N) {
    int i = blockIdx.x * blockDim.x + threadIdx.x;
    if (i >= N) return;
    const float4* hr4 = (const float4*)(h + (long)i * 32);
    float hv[32];
#pragma unroll
    for (int q = 0; q < 8; ++q) {
        float4 v = hr4[q];
        hv[q * 4 + 0] = v.x; hv[q * 4 + 1] = v.y;
        hv[q * 4 + 2] = v.z; hv[q * 4 + 3] = v.w;
    }
    const long b = batch[i];
    for (int c = 0; c < 10; ++c) {
        float acc = bl0[c];
#pragma unroll
        for (int f = 0; f < 32; ++f) acc += hv[f] * Wl0[f * 10 + c];
        atomicMax(&pool0[b * 16 + c], encf(acc));
    }
}

// ---------------- degree / normalization ----------------------------------
__global__ void deg_accum_kernel(const int* __restrict__ col,
                                 const int* __restrict__ ew,
                                 float* __restrict__ deg, int E) {
    int e = blockIdx.x * blockDim.x + threadIdx.x;
    if (e >= E) return;
    if (ew[e] == 1) atomicAdd(&deg[col[e]], 1.0f);
}
__global__ void dinv_kernel(const float* __restrict__ deg,
                            float* __restrict__ dinv, int N) {
    int i = blockIdx.x * blockDim.x + threadIdx.x;
    if (i < N) dinv[i] = rsqrtf(deg[i]);
}

// ---------------- GCN aggregation (F == 64) -------------------------------
__global__ void agg_init_kernel(const float* __restrict__ hg,
                                const float* __restrict__ dinv,
                                float* __restrict__ agg, long total) {
    long t = (long)blockIdx.x * blockDim.x + threadIdx.x;
    if (t >= total) return;
    float di = dinv[t >> 6];
    agg[t] = hg[t] * di * di;   // self-loop term h * dinv^2
}
__global__ void agg_edges_kernel(const int* __restrict__ row,
                                 const int* __restrict__ col,
                                 const int* __restrict__ ew,
                                 const float* __restrict__ dinv,
                                 const float* __restrict__ hg,
                                 float* __restrict__ agg, int E) {
    long t = (long)blockIdx.x * blockDim.x + threadIdx.x;
    long e = t >> 2;              // 4 threads per edge, 16 floats each
    if (e >= E) return;
    if (ew[e] != 1) return;
    const int p = (int)(t & 3);
    const int r = row[e], c = col[e];
    const float s = dinv[r] * dinv[c];
    const float4* __restrict__ src = (const float4*)(hg + ((long)r << 6) + (p << 4));
    float* __restrict__        dst = agg + ((long)c << 6) + (p << 4);
#pragma unroll
    for (int q = 0; q < 4; ++q) {
        float4 v = src[q];
        atomicAdd(dst + q * 4 + 0, v.x * s);
        atomicAdd(dst + q * 4 + 1, v.y * s);
        atomicAdd(dst + q * 4 + 2, v.z * s);
        atomicAdd(dst + q * 4 + 3, v.w * s);
    }
}

// ---------------- bias add + segment-max pool (F == 64) -------------------
__global__ void bias_pool_kernel(float* __restrict__ hb,
                                 const float* __restrict__ bias,
                                 const int* __restrict__ batch,
                                 unsigned* __restrict__ pool, long total) {
    long t = (long)blockIdx.x * blockDim.x + threadIdx.x;
    if (t >= total) return;
    const int i = (int)(t >> 6);
    const int f = (int)(t & 63);
    float v = hb[t] + bias[f];
    hb[t] = v;
    atomicMax(&pool[((long)batch[i] << 6) + f], encf(v));
}

// ---------------- final readout: out[G,10] --------------------------------
__global__ void final_out_kernel(const unsigned* __restrict__ pool0,
                                 const unsigned* __restrict__ pool1,
                                 const unsigned* __restrict__ pool2,
                                 const float* __restrict__ Wl1,
                                 const float* __restrict__ bl1,
                                 const float* __restrict__ Wl2,
                                 const float* __restrict__ bl2,
                                 float* __restrict__ out, int G) {
    int t = blockIdx.x * blockDim.x + threadIdx.x;
    if (t >= G * 10) return;
    const int g = t / 10, c = t % 10;
    float acc = decf(pool0[(long)g * 16 + c]);
    const unsigned* p1 = pool1 + ((long)g << 6);
    const unsigned* p2 = pool2 + ((long)g << 6);
    for (int f = 0; f < 64; ++f) acc += decf(p1[f]) * Wl1[f * 10 + c];
    acc += bl1[c];
    for (int f = 0; f < 64; ++f) acc += decf(p2[f]) * Wl2[f * 10 + c];
    acc += bl2[c];
    out[t] = acc;
}

static inline int cdiv(long a, long b) { return (int)((a + b - 1) / b); }

extern "C" void kernel_launch(void* const* d_in, const int* in_sizes, int n_in,
                              void* d_out, int out_size, void* d_ws, size_t ws_size,
                              hipStream_t stream) {
    const float* x   = (const float*)d_in[0];
    const int* eidx  = (const int*)d_in[1];
    const int* ew    = (const int*)d_in[2];
    const int* batch = (const int*)d_in[3];
    const float* W1  = (const float*)d_in[5];
    const float* b1  = (const float*)d_in[6];
    const float* g1  = (const float*)d_in[7];
    const float* bt1 = (const float*)d_in[8];
    const float* W2  = (const float*)d_in[9];
    const float* b2  = (const float*)d_in[10];
    const float* g2  = (const float*)d_in[11];
    const float* bt2 = (const float*)d_in[12];
    const float* Wl0 = (const float*)d_in[13];
    const float* bl0 = (const float*)d_in[14];
    const float* Wg1 = (const float*)d_in[15];
    const float* bg1 = (const float*)d_in[16];
    const float* Wl1 = (const float*)d_in[17];
    const float* bl1 = (const float*)d_in[18];
    const float* Wg2 = (const float*)d_in[19];
    const float* bg2 = (const float*)d_in[20];
    const float* Wl2 = (const float*)d_in[21];
    const float* bl2 = (const float*)d_in[22];
    float* out = (float*)d_out;

    const int N = in_sizes[0] / 128;
    const int E = in_sizes[2];
    const int G = out_size / 10;       // num_graphs lives on device; C == 10
    const int* row = eidx;
    const int* col = eidx + E;

    // ---- workspace layout (floats) ----
    float* ws = (float*)d_ws;
    size_t off = 0;
    float* bufA = ws + off; off += (size_t)N * 64;   // z1/h1, later agg1/h_gcn1
    float* bufB = ws + off; off += (size_t)N * 64;   // z2/h2, later agg2/h_gcn2
    float* bufC = ws + off; off += (size_t)N * 64;   // hg1, later hg2
    float* deg   = ws + off; off += N;
    float* dinv  = ws + off; off += N;
    float* stats = ws + off; off += 128;             // [sum1,sq1 | sum2,sq2]
    float* murs  = ws + off; off += 128;             // [mu1,rsig1 | mu2,rsig2]
    unsigned* pool0 = (unsigned*)(ws + off); off += (size_t)G * 16;
    unsigned* pool1 = (unsigned*)(ws + off); off += (size_t)G * 64;
    unsigned* pool2 = (unsigned*)(ws + off); off += (size_t)G * 64;
    (void)ws_size; (void)n_in;

    const int B = 256;
    const int tiles = (N + 15) / 16;
    const int gemmBlocks = (tiles + 7) / 8;          // 8 waves / block
    const float invN = 1.0f / (float)N;

    // ---- init ----
    fill_f32_kernel<<<cdiv(N, B), B, 0, stream>>>(deg, N, 1.0f);
    fill_f32_kernel<<<1, B, 0, stream>>>(stats, 128, 0.0f);
    fill_u32_kernel<<<cdiv((long)G * 144, B), B, 0, stream>>>(pool0, (long)G * 144, ENC_NINF);

    // ---- initial MLP ----
    gemm_wmma_kernel<128, 32><<<gemmBlocks, B, (128 + 128 * 32) * sizeof(float), stream>>>(
        x, W1, b1, bufA, stats, N);
    finalize_stats_kernel<<<1, 32, 0, stream>>>(stats, murs, invN);
    bn_relu_kernel<<<cdiv((long)N * 32, B), B, 0, stream>>>(bufA, murs, g1, bt1, (long)N * 32);

    gemm_wmma_kernel<32, 32><<<gemmBlocks, B, (128 + 32 * 32) * sizeof(float), stream>>>(
        bufA, W2, b2, bufB, stats + 64, N);
    finalize_stats_kernel<<<1, 32, 0, stream>>>(stats + 64, murs + 64, invN);
    bn_relu_kernel<<<cdiv((long)N * 32, B), B, 0, stream>>>(bufB, murs + 64, g2, bt2, (long)N * 32);

    // ---- initial_linear + pool0 ----
    pool0_kernel<<<cdiv(N, B), B, 0, stream>>>(bufB, Wl0, bl0, batch, pool0, N);

    // ---- degrees / norm ----
    deg_accum_kernel<<<cdiv(E, B), B, 0, stream>>>(col, ew, deg, E);
    dinv_kernel<<<cdiv(N, B), B, 0, stream>>>(deg, dinv, N);

    // ---- GCN layer 1: hg1 = h2 @ Wg1 ; agg ; +bias ; pool1 ----
    gemm_wmma_kernel<32, 64><<<gemmBlocks, B, (128 + 32 * 64) * sizeof(float), stream>>>(
        bufB, Wg1, nullptr, bufC, nullptr, N);
    agg_init_kernel<<<cdiv((long)N * 64, B), B, 0, stream>>>(bufC, dinv, bufA, (long)N * 64);
    agg_edges_kernel<<<cdiv((long)E * 4, B), B, 0, stream>>>(row, col, ew, dinv, bufC, bufA, E);
    bias_pool_kernel<<<cdiv((long)N * 64, B), B, 0, stream>>>(bufA, bg1, batch, pool1, (long)N * 64);

    // ---- GCN layer 2: hg2 = h_gcn1 @ Wg2 ; agg ; +bias ; pool2 ----
    gemm_wmma_kernel<64, 64><<<gemmBlocks, B, (128 + 64 * 64) * sizeof(float), stream>>>(
        bufA, Wg2, nullptr, bufC, nullptr, N);
    agg_init_kernel<<<cdiv((long)N * 64, B), B, 0, stream>>>(bufC, dinv, bufB, (long)N * 64);
    agg_edges_kernel<<<cdiv((long)E * 4, B), B, 0, stream>>>(row, col, ew, dinv, bufC, bufB, E);
    bias_pool_kernel<<<cdiv((long)N * 64, B), B, 0, stream>>>(bufB, bg2, batch, pool2, (long)N * 64);

    // ---- final readout ----
    final_out_kernel<<<cdiv((long)G * 10, B), B, 0, stream>>>(
        pool0, pool1, pool2, Wl1, bl1, Wl2, bl2, out, G);
}